// Embedding_11338713661676
// MI455X (gfx1250) — compile-verified
//
#include <hip/hip_runtime.h>
#include <hip/hip_bf16.h>
#include <stdint.h>

// Embedding gather: out[t, :] = w[x[t], :]
//   x: [B*S] int32 (16384 tokens), w: [32000, 1024] f32, out: [B*S, 1024] f32
// Pure data movement (~134 MB min traffic -> ~5.7us @ 23.3 TB/s).
// Use the CDNA5 Tensor Data Mover in gather mode: one D# gathers 8 rows into
// LDS, a second D# streams the compacted 32KB tile contiguously to out.

#define EMB_DIM 1024
#define EMB_VOCAB 32000
#define ROWS_PER_BLOCK 8

typedef unsigned int v4u __attribute__((ext_vector_type(4)));
typedef int          v8i __attribute__((ext_vector_type(8)));
typedef int          v4i __attribute__((ext_vector_type(4)));
typedef float        v4f __attribute__((ext_vector_type(4)));

#if defined(__gfx1250__) && __has_builtin(__builtin_amdgcn_tensor_load_to_lds) && \
    __has_builtin(__builtin_amdgcn_tensor_store_from_lds)
#define USE_TDM 1
#else
#define USE_TDM 0
#endif

__device__ __forceinline__ void wait_tensorcnt0() {
#if defined(__gfx1250__)
#if __has_builtin(__builtin_amdgcn_s_wait_tensorcnt)
  __builtin_amdgcn_s_wait_tensorcnt(0);
#else
  asm volatile("s_wait_tensorcnt 0x0" ::: "memory");
#endif
#endif
}

__global__ __launch_bounds__(32) void embed_gather_tdm(
    const int* __restrict__ x, const float* __restrict__ w,
    float* __restrict__ out, int n_tokens) {
#if USE_TDM
  __shared__ float lds_buf[ROWS_PER_BLOCK * EMB_DIM]; // 32 KB tile

  const int t0 = blockIdx.x * ROWS_PER_BLOCK;
  int nr = n_tokens - t0;
  if (nr <= 0) return;
  if (nr > ROWS_PER_BLOCK) nr = ROWS_PER_BLOCK;

  const uint32_t lds_off = (uint32_t)(uintptr_t)(&lds_buf[0]);
  const uint64_t w_addr  = (uint64_t)(uintptr_t)w;
  const uint64_t o_addr  = (uint64_t)(uintptr_t)(out + (size_t)t0 * EMB_DIM);

  // ---------------- Load descriptor: gather nr rows of w into LDS ----------
  // Group0: count=1 | gather_index_size=1 (32-bit) << 30 | gather_mode=1 << 31
  v4u g0l;
  g0l[0] = 0xC0000001u;
  g0l[1] = lds_off;                                   // lds_addr (bytes)
  g0l[2] = (uint32_t)(w_addr & 0xFFFFFFFFull);        // global_addr[31:0]
  g0l[3] = (uint32_t)((w_addr >> 32) & 0x01FFFFFFull) // global_addr[56:32]
           | (2u << 30);                              // type = 2 ("image")
  // Group1
  v8i g1l;
  g1l[0] = 0x00020000;                        // data_size = 2 (4 bytes)
  g1l[1] = (int)((EMB_DIM & 0xFFFF) << 16);   // tensor_dim0[15:0] = 1024
  g1l[2] = (int)((EMB_VOCAB & 0xFFFF) << 16); // tensor_dim1[15:0] = 32000
  g1l[3] = (int)((EMB_DIM & 0xFFFF) << 16);   // tile_dim0 = 1024 (full row)
  g1l[4] = nr;                                // tile_dim1 = #valid indices
  g1l[5] = EMB_DIM;                           // tensor_dim0_stride = 1024
  g1l[6] = 0;
  g1l[7] = 0;
  // Groups 2/3: 32-bit row indices (uniform scalar loads; clamp the tail,
  // duplicate indices are legal in gather mode).
  v4i g2l, g3l;
  {
    int last = n_tokens - 1;
    int i0 = t0 + 0 > last ? last : t0 + 0;
    int i1 = t0 + 1 > last ? last : t0 + 1;
    int i2 = t0 + 2 > last ? last : t0 + 2;
    int i3 = t0 + 3 > last ? last : t0 + 3;
    int i4 = t0 + 4 > last ? last : t0 + 4;
    int i5 = t0 + 5 > last ? last : t0 + 5;
    int i6 = t0 + 6 > last ? last : t0 + 6;
    int i7 = t0 + 7 > last ? last : t0 + 7;
    g2l[0] = x[i0]; g2l[1] = x[i1]; g2l[2] = x[i2]; g2l[3] = x[i3];
    g3l[0] = x[i4]; g3l[1] = x[i5]; g3l[2] = x[i6]; g3l[3] = x[i7];
  }

  // --------------- Store descriptor: contiguous nr*1024 f32 LDS -> out -----
  const int n = nr * EMB_DIM; // <= 8192, fits tile_dim0's 16 bits
  v4u g0s;
  g0s[0] = 1u; // count=1, normal mode
  g0s[1] = lds_off;
  g0s[2] = (uint32_t)(o_addr & 0xFFFFFFFFull);
  g0s[3] = (uint32_t)((o_addr >> 32) & 0x01FFFFFFull) | (2u << 30);
  v8i g1s;
  g1s[0] = 0x00020000;                   // data_size = 4 bytes
  g1s[1] = (int)((n & 0xFFFF) << 16);    // tensor_dim0 = n
  g1s[2] = (int)((1 & 0xFFFF) << 16);    // tensor_dim1 = 1
  g1s[3] = (int)((n & 0xFFFF) << 16);    // tile_dim0 = n (1-D tile)
  g1s[4] = 0;                            // tile_dim1/2 unused
  g1s[5] = n;                            // tensor_dim0_stride
  g1s[6] = 0;
  g1s[7] = 0;
  v4i z4 = {0, 0, 0, 0};

#if __clang_major__ >= 23
  v8i z8 = {0, 0, 0, 0, 0, 0, 0, 0};
  __builtin_amdgcn_tensor_load_to_lds(g0l, g1l, g2l, g3l, z8, 0);
  wait_tensorcnt0();
  __builtin_amdgcn_tensor_store_from_lds(g0s, g1s, z4, z4, z8, 0);
#else
  __builtin_amdgcn_tensor_load_to_lds(g0l, g1l, g2l, g3l, 0);
  wait_tensorcnt0();
  __builtin_amdgcn_tensor_store_from_lds(g0s, g1s, z4, z4, 0);
#endif
  wait_tensorcnt0();
#else
  // Fallback: coalesced B128 gather; NT stores keep w resident in the 192MB L2.
  const int t0 = blockIdx.x * ROWS_PER_BLOCK;
  for (int r = 0; r < ROWS_PER_BLOCK; ++r) {
    int t = t0 + r;
    if (t >= n_tokens) break;
    int row = x[t];
    const v4f* __restrict__ src = (const v4f*)(w + (size_t)row * EMB_DIM);
    v4f* __restrict__ dst = (v4f*)(out + (size_t)t * EMB_DIM);
    for (int i = threadIdx.x; i < EMB_DIM / 4; i += 32) {
      __builtin_nontemporal_store(src[i], &dst[i]);
    }
  }
#endif
}

extern "C" void kernel_launch(void* const* d_in, const int* in_sizes, int n_in,
                              void* d_out, int out_size, void* d_ws, size_t ws_size,
                              hipStream_t stream) {
  const int*   x   = (const int*)d_in[0];   // [B*S] token ids
  const float* w   = (const float*)d_in[1]; // [VOCAB, DIM]
  float*       out = (float*)d_out;         // [B*S, DIM]
  const int n_tokens = in_sizes[0];          // 16384

  const int blocks = (n_tokens + ROWS_PER_BLOCK - 1) / ROWS_PER_BLOCK; // 2048
  embed_gather_tdm<<<blocks, 32, 0, stream>>>(x, w, out, n_tokens);
}